// OGProjection_65601330479437
// MI455X (gfx1250) — compile-verified
//
#include <hip/hip_runtime.h>

// Problem constants from the reference (B,C,H,W,D fixed by the harness).
#define BB 4
#define CC 8
#define HH 160
#define WW 160
#define DD 160
#define BC (BB * CC)          // 32 == one wave32: lane = b*8 + c
#define HWD (HH * WW * DD)    // elements per channel
#define WAVES_PER_BLOCK 8

// One wave per output row n. lane = b*8+c; each lane gathers the 8 trilinear
// corners of channel c of image b at coords[b*n_per + n] and writes one f32
// of the contiguous 128B output row. Gather-latency-bound: all 8 corner loads
// are issued before any use so one s_wait_loadcnt covers the whole fetch.
__global__ __launch_bounds__(256)
void OGProjection_trilerp_kernel(const float* __restrict__ img,
                                 const float* __restrict__ gf,
                                 const int*   __restrict__ image_dim_p,
                                 float*       __restrict__ out,
                                 int n_per)
{
    const int lane = threadIdx.x & 31;
    const int row  = blockIdx.x * WAVES_PER_BLOCK + (threadIdx.x >> 5);
    if (row >= n_per) return;

    const int b = lane >> 3;   // image index 0..3
    const int c = lane & 7;    // channel index 0..7

    // factor = image_dim / H (wave-uniform; ==1.0 for this problem instance).
    const float factor = (float)(*image_dim_p) / (float)HH;

    // Coordinate fetch for (b, row); rows are consecutive across waves on the
    // same WGP, so prefetch ahead at WGP scope (pull into all cache levels).
    const float* __restrict__ g = gf + (size_t)(b * n_per + row) * 3;
    {
        int pref = row + 64; pref = pref < n_per ? pref : (n_per - 1);
        __builtin_prefetch(gf + (size_t)(b * n_per + pref) * 3, 0, 3);
    }
    float x = g[0];
    float y = g[1];
    float z = g[2];
    // Uniform fast path: division by exactly 1.0 is the identity (bit-exact),
    // so skip the ~30-VALU IEEE divide expansion when factor == 1.0f.
    if (factor != 1.0f) {
        x = x / factor;
        y = y / factor;
        z = z / factor;
    }

    // bounds(v, size): lo = min(floor(v), size-1), hi = min(ceil(v), size-1)
    const float Hm1 = (float)(HH - 1);
    const float Wm1 = (float)(WW - 1);
    const float Dm1 = (float)(DD - 1);
    const float x1f = fminf(floorf(x), Hm1), x2f = fminf(ceilf(x), Hm1);
    const float y1f = fminf(floorf(y), Wm1), y2f = fminf(ceilf(y), Wm1);
    const float z1f = fminf(floorf(z), Dm1), z2f = fminf(ceilf(z), Dm1);

    const int x1 = (int)x1f, x2 = (int)x2f;
    const int y1 = (int)y1f, y2 = (int)y2f;
    const int z1 = (int)z1f, z2 = (int)z2f;

    // Reference weights (degenerate-to-zero when floor==ceil, kept faithfully).
    const float wx = x - x1f, wx2 = x2f - x;
    const float wy = y - y1f, wy2 = y2f - y;
    const float wz = z - z1f, wz2 = z2f - z;

    // 32-bit element offsets: total tensor is 131M elements (<2^31 bytes),
    // so the compiler keeps `img` as an SGPR base + per-lane 32-bit voffset.
    const int chan = (b * CC + c) * HWD;
    const int rx1 = x1 * (WW * DD), rx2 = x2 * (WW * DD);
    const int ry1 = y1 * DD,        ry2 = y2 * DD;
    const float* __restrict__ p = img + chan;

    // Issue all 8 corner gathers before any arithmetic (deep MLP per lane).
    const float v111 = p[rx1 + ry1 + z1];
    const float v211 = p[rx2 + ry1 + z1];
    const float v121 = p[rx1 + ry2 + z1];
    const float v221 = p[rx2 + ry2 + z1];
    const float v112 = p[rx1 + ry1 + z2];
    const float v212 = p[rx2 + ry1 + z2];
    const float v122 = p[rx1 + ry2 + z2];
    const float v222 = p[rx2 + ry2 + z2];

    // z == z1 slab
    const float lx1a = v211 * wx + v111 * wx2;
    const float lx2a = v221 * wx + v121 * wx2;
    const float la   = lx2a * wy + lx1a * wy2;
    // z == z2 slab
    const float lx1b = v212 * wx + v112 * wx2;
    const float lx2b = v222 * wx + v122 * wx2;
    const float lb   = lx2b * wy + lx1b * wy2;

    const float r = lb * wz + la * wz2;

    // out[n, b*8+c]: one contiguous 128B store per wave.
    out[(size_t)row * BC + lane] = r;
}

extern "C" void kernel_launch(void* const* d_in, const int* in_sizes, int n_in,
                              void* d_out, int out_size, void* d_ws, size_t ws_size,
                              hipStream_t stream) {
    (void)n_in; (void)out_size; (void)d_ws; (void)ws_size;
    const float* img       = (const float*)d_in[0];   // [B,C,H,W,D] f32
    const float* gf        = (const float*)d_in[1];   // [B*n_per, 3] f32
    // d_in[2] = batch (numerically unused by the reference)
    const int*   image_dim = (const int*)d_in[3];     // scalar int

    const int n_total = in_sizes[2];     // B * n_per
    const int n_per   = n_total / BB;

    const int blocks = (n_per + WAVES_PER_BLOCK - 1) / WAVES_PER_BLOCK;
    OGProjection_trilerp_kernel<<<blocks, WAVES_PER_BLOCK * 32, 0, stream>>>(
        img, gf, image_dim, (float*)d_out, n_per);
}